// SimpleGNN_4784593568475
// MI455X (gfx1250) — compile-verified
//
#include <hip/hip_runtime.h>
#include <hip/hip_bf16.h>

typedef __attribute__((ext_vector_type(2))) float v2f;
typedef __attribute__((ext_vector_type(8))) float v8f;

#define N_NODES 100000
#define N_EDGES 6400000
#define N_TILES (N_NODES / 16)   // 6250 exactly
#define ETOT (N_EDGES + N_NODES) // edges + self loops

// ---------------------------------------------------------------- degree
__global__ __launch_bounds__(256) void gcn_deg_kernel(const int* __restrict__ dst,
                                                      float* __restrict__ deg) {
    int e = blockIdx.x * blockDim.x + threadIdx.x;
    if (e < N_EDGES) atomicAdd(&deg[dst[e]], 1.0f);
}

__global__ __launch_bounds__(256) void gcn_dinv_kernel(float* __restrict__ deg_to_dinv) {
    int n = blockIdx.x * blockDim.x + threadIdx.x;
    if (n < N_NODES) {
        float d = deg_to_dinv[n] + 1.0f;   // +1 self loop; always > 0
        deg_to_dinv[n] = rsqrtf(d);
    }
}

// ---------------------------------------------------------------- layer 1 linear: h1 = x @ W1  (K=10 padded, N=16)
__global__ __launch_bounds__(256) void gcn_linear1_wmma(const float* __restrict__ x,
                                                        const float* __restrict__ W1,
                                                        float* __restrict__ h1) {
    int wave = (blockIdx.x * (blockDim.x >> 5)) + (threadIdx.x >> 5); // tile id (wave-uniform)
    if (wave >= N_TILES) return;                                      // uniform exit: EXEC stays all-1
    int lane  = threadIdx.x & 31;
    int m     = lane & 15;              // row (M) for A, col (N) for B
    int khalf = (lane >> 4) * 2;        // lanes 16..31 hold K+2,K+3
    const float* xrow = x + (wave * 16 + m) * 10;

    v8f acc = {};
    #pragma unroll
    for (int k0 = 0; k0 < 12; k0 += 4) {
        int ka = k0 + khalf;
        v2f a, b;
        a.x = (ka     < 10) ? xrow[ka]     : 0.0f;
        a.y = (ka + 1 < 10) ? xrow[ka + 1] : 0.0f;
        b.x = (ka     < 10) ? W1[ka * 16 + m]       : 0.0f;
        b.y = (ka + 1 < 10) ? W1[(ka + 1) * 16 + m] : 0.0f;
        acc = __builtin_amdgcn_wmma_f32_16x16x4_f32(false, a, false, b,
                                                    (short)0, acc, false, false);
    }
    // D layout: VGPR r -> rows r (lanes 0-15) and r+8 (lanes 16-31), col = lane&15
    int mbase = (lane >> 4) * 8;
    #pragma unroll
    for (int r = 0; r < 8; ++r)
        h1[(wave * 16 + mbase + r) * 16 + m] = acc[r];
}

// ---------------------------------------------------------------- layer 1 aggregation (lane = feature)
__global__ __launch_bounds__(256) void gcn_agg1_kernel(const int* __restrict__ src,
                                                       const int* __restrict__ dst,
                                                       const float* __restrict__ dinv,
                                                       const float* __restrict__ h1,
                                                       float* __restrict__ agg1) {
    long long idx = (long long)blockIdx.x * blockDim.x + threadIdx.x;
    long long e = idx >> 4;
    int f = (int)(idx & 15);
    if (e >= (long long)ETOT) return;
    int s, d;
    if (e < N_EDGES) { s = src[e]; d = dst[e]; }
    else             { s = d = (int)(e - N_EDGES); }  // self loop
    float nrm = dinv[s] * dinv[d];
    atomicAdd(&agg1[d * 16 + f], h1[s * 16 + f] * nrm);
}

// ---------------------------------------------------------------- layer 2 linear: h2 = relu(agg1 + b1) @ W2 (K=16, N=2)
__global__ __launch_bounds__(256) void gcn_linear2_wmma(const float* __restrict__ agg1,
                                                        const float* __restrict__ b1,
                                                        const float* __restrict__ W2,
                                                        float* __restrict__ h2) {
    int wave = (blockIdx.x * (blockDim.x >> 5)) + (threadIdx.x >> 5);
    if (wave >= N_TILES) return;
    int lane  = threadIdx.x & 31;
    int m     = lane & 15;
    int khalf = (lane >> 4) * 2;
    const float* arow = agg1 + (wave * 16 + m) * 16;

    v8f acc = {};
    #pragma unroll
    for (int k0 = 0; k0 < 16; k0 += 4) {
        int ka = k0 + khalf;
        v2f a, b;
        float a0 = arow[ka]     + b1[ka];
        float a1 = arow[ka + 1] + b1[ka + 1];
        a.x = a0 > 0.0f ? a0 : 0.0f;     // relu
        a.y = a1 > 0.0f ? a1 : 0.0f;
        b.x = (m < 2) ? W2[ka * 2 + m]       : 0.0f;   // W2 is 16x2, pad N to 16
        b.y = (m < 2) ? W2[(ka + 1) * 2 + m] : 0.0f;
        acc = __builtin_amdgcn_wmma_f32_16x16x4_f32(false, a, false, b,
                                                    (short)0, acc, false, false);
    }
    int mbase = (lane >> 4) * 8;
    if (m < 2) {
        #pragma unroll
        for (int r = 0; r < 8; ++r)
            h2[(wave * 16 + mbase + r) * 2 + m] = acc[r];
    }
}

// ---------------------------------------------------------------- output init with bias
__global__ __launch_bounds__(256) void gcn_init_out_kernel(const float* __restrict__ b2,
                                                           float* __restrict__ out) {
    int idx = blockIdx.x * blockDim.x + threadIdx.x;
    if (idx < N_NODES * 2) out[idx] = b2[idx & 1];
}

// ---------------------------------------------------------------- layer 2 aggregation (lane pair = 2 outputs)
__global__ __launch_bounds__(256) void gcn_agg2_kernel(const int* __restrict__ src,
                                                       const int* __restrict__ dst,
                                                       const float* __restrict__ dinv,
                                                       const float* __restrict__ h2,
                                                       float* __restrict__ out) {
    long long idx = (long long)blockIdx.x * blockDim.x + threadIdx.x;
    long long e = idx >> 1;
    int c = (int)(idx & 1);
    if (e >= (long long)ETOT) return;
    int s, d;
    if (e < N_EDGES) { s = src[e]; d = dst[e]; }
    else             { s = d = (int)(e - N_EDGES); }
    float nrm = dinv[s] * dinv[d];
    atomicAdd(&out[d * 2 + c], h2[s * 2 + c] * nrm);
}

extern "C" void kernel_launch(void* const* d_in, const int* in_sizes, int n_in,
                              void* d_out, int out_size, void* d_ws, size_t ws_size,
                              hipStream_t stream) {
    const float* x   = (const float*)d_in[0];
    const int*   ei  = (const int*)d_in[1];      // [2, E] flattened: row 0 = src, row 1 = dst
    const float* W1  = (const float*)d_in[2];
    const float* b1  = (const float*)d_in[3];
    const float* W2  = (const float*)d_in[4];
    const float* b2  = (const float*)d_in[5];
    float*       out = (float*)d_out;

    const int* src = ei;
    const int* dst = ei + N_EDGES;

    // workspace layout: dinv[N] | h1[16N] | agg1[16N] | h2[2N]  (~14 MB)
    float* ws   = (float*)d_ws;
    float* dinv = ws;
    float* h1   = ws + N_NODES;
    float* agg1 = ws + N_NODES + 16 * N_NODES;
    float* h2   = ws + N_NODES + 32 * N_NODES;

    hipMemsetAsync(dinv, 0, (size_t)N_NODES * sizeof(float), stream);
    hipMemsetAsync(agg1, 0, (size_t)N_NODES * 16 * sizeof(float), stream);

    // degree + normalization
    gcn_deg_kernel<<<(N_EDGES + 255) / 256, 256, 0, stream>>>(dst, dinv);
    gcn_dinv_kernel<<<(N_NODES + 255) / 256, 256, 0, stream>>>(dinv);

    // layer 1: linear (WMMA) then scatter-add aggregation
    {
        int waves_per_block = 256 / 32;
        int grid = (N_TILES + waves_per_block - 1) / waves_per_block;
        gcn_linear1_wmma<<<grid, 256, 0, stream>>>(x, W1, h1);
    }
    {
        long long total = (long long)ETOT * 16;
        int grid = (int)((total + 255) / 256);
        gcn_agg1_kernel<<<grid, 256, 0, stream>>>(src, dst, dinv, h1, agg1);
    }

    // layer 2: relu + bias + linear (WMMA), then aggregation into out (pre-seeded with b2)
    {
        int waves_per_block = 256 / 32;
        int grid = (N_TILES + waves_per_block - 1) / waves_per_block;
        gcn_linear2_wmma<<<grid, 256, 0, stream>>>(agg1, b1, W2, h2);
    }
    gcn_init_out_kernel<<<(N_NODES * 2 + 255) / 256, 256, 0, stream>>>(b2, out);
    {
        long long total = (long long)ETOT * 2;
        int grid = (int)((total + 255) / 256);
        gcn_agg2_kernel<<<grid, 256, 0, stream>>>(src, dst, dinv, h2, out);
    }
}